// HGCNForTextClassification_40879498723409
// MI455X (gfx1250) — compile-verified
//
#include <hip/hip_runtime.h>
#include <hip/hip_bf16.h>
#include <math.h>

// ---------------------------------------------------------------------------
// HGCN (HAN-style) forward for MI455X / gfx1250.
// fp32 end-to-end; GEMMs use V_WMMA_F32_16X16X4_F32 (wave32) with
// double-buffered async global->LDS staging of the A tiles (ASYNCcnt).
// ---------------------------------------------------------------------------

typedef __attribute__((ext_vector_type(2))) float v2f;
typedef __attribute__((ext_vector_type(8))) float v8f;
typedef int hgcn_v4i __attribute__((vector_size(16)));   // matches builtin param

#define DIMD 128
#define BPAD 132   // LDS row stride (floats): 132 mod 64 == 4 -> conflict-free frags

// ---- gfx1250 async global->LDS (B128) with portable fallback --------------
#if defined(__has_builtin)
#if __has_builtin(__builtin_amdgcn_global_load_async_to_lds_b128)
#define HGCN_HAVE_ASYNC_LDS 1
#endif
#endif

__device__ __forceinline__ void hgcn_copy16_to_lds(const float* __restrict__ g,
                                                   float* __restrict__ l) {
#if defined(HGCN_HAVE_ASYNC_LDS)
    __builtin_amdgcn_global_load_async_to_lds_b128(
        (__attribute__((address_space(1))) hgcn_v4i*)g,
        (__attribute__((address_space(3))) hgcn_v4i*)l,
        /*offset=*/0, /*cpol=*/0);
#else
    *(float4*)l = *(const float4*)g;
#endif
}

__device__ __forceinline__ void hgcn_wait_async() {
#if defined(HGCN_HAVE_ASYNC_LDS)
#if __has_builtin(__builtin_amdgcn_s_wait_asynccnt)
    __builtin_amdgcn_s_wait_asynccnt(0);
#else
    asm volatile("s_wait_asynccnt 0x0" ::: "memory");
#endif
#endif
}

// --------------------------- WMMA GEMM: C[N,128] = A[N,128] @ B[128,128] ----
// Block: 256 threads (8 waves). Wave w owns output cols [16w, 16w+16).
// B staged transposed+padded in LDS once per block; 16x128 A tiles staged
// via double-buffered async B128 copies. K-loop: 32 x v_wmma_f32_16x16x4_f32.
__global__ __launch_bounds__(256)
void hgcn_gemm_n128(const float* __restrict__ A, const float* __restrict__ Bm,
                    float* __restrict__ C, int nRows) {
    __shared__ float sB[DIMD * BPAD];      // sB[col*BPAD + k] = B[k][col]
    __shared__ float sA[2][16 * BPAD];     // double-buffered A tiles

    const int tid = threadIdx.x;
    for (int i = tid; i < DIMD * DIMD; i += 256) {
        int k = i >> 7, col = i & 127;
        sB[col * BPAD + k] = Bm[i];        // coalesced global read, transposed store
    }

    const int wave = tid >> 5;
    const int lane = tid & 31;
    const int m    = lane & 15;            // A-row / B-col index within tile
    const int hi   = lane >> 4;            // half-wave selects K pair / C row half
    const int colBase = wave * 16;
    const int nTiles  = (nRows + 15) >> 4;

    // Stage one 16x128 A tile: 512 float4 chunks; thread t moves chunks t, t+256.
    auto stageA = [&](int tile, int buf) {
        const int row0 = tile * 16;
        #pragma unroll
        for (int j = 0; j < 2; ++j) {
            int c = tid + j * 256;
            int r = c >> 5, q = c & 31;    // row in tile, float4 index in row
            int row = row0 + r;
            float* lp = &sA[buf][r * BPAD + q * 4];
            if (row < nRows) {
                hgcn_copy16_to_lds(&A[(size_t)row * DIMD + q * 4], lp);
            } else {
                float4 z = {0.f, 0.f, 0.f, 0.f};
                *(float4*)lp = z;
            }
        }
    };

    int tile = blockIdx.x;
    if (tile < nTiles) stageA(tile, 0);
    hgcn_wait_async();
    __syncthreads();                       // sB + first A tile visible

    int buf = 0;
    for (; tile < nTiles; tile += gridDim.x) {
        const int tnext = tile + gridDim.x;
        if (tnext < nTiles) stageA(tnext, buf ^ 1);   // overlap with compute

        v8f acc = {};
        #pragma unroll
        for (int k0 = 0; k0 < DIMD; k0 += 4) {
            // A frag (16x4): lane m, v0/v1 = K = k0+2*hi, k0+2*hi+1
            const float* pa = &sA[buf][m * BPAD + k0 + 2 * hi];
            v2f a; a.x = pa[0]; a.y = pa[1];
            // B frag (4x16): lane n=m, same K split per half-wave
            const float* pb = &sB[(colBase + m) * BPAD + k0 + 2 * hi];
            v2f b; b.x = pb[0]; b.y = pb[1];
            acc = __builtin_amdgcn_wmma_f32_16x16x4_f32(
                false, a, false, b, (short)0, acc, false, false);
        }

        // C 16x16 f32 layout: VGPR v -> row v + 8*hi, lane n = m
        const int row0 = tile * 16;
        float* pc = C + (size_t)row0 * DIMD + colBase;
        #pragma unroll
        for (int v = 0; v < 8; ++v) {
            int row = v + 8 * hi;
            if (row0 + row < nRows) pc[(size_t)row * DIMD + m] = acc[v];
        }

        hgcn_wait_async();                 // next tile landed in alternate buffer
        __syncthreads();                   // all waves done with sA[buf] + loads done
        buf ^= 1;
    }
}

// --------------------------- degree / normalization -------------------------
__global__ void hgcn_init_deg(float* deg, int n) {
    int i = blockIdx.x * 256 + threadIdx.x;
    if (i < n) deg[i] = 1.0f;                     // self-loop contribution
}
__global__ void hgcn_count_deg(const int* __restrict__ dst, float* deg, int E) {
    int e = blockIdx.x * 256 + threadIdx.x;
    if (e < E) atomicAdd(&deg[dst[e]], 1.0f);
}
__global__ void hgcn_rsqrt_deg(float* deg, int n) {
    int i = blockIdx.x * 256 + threadIdx.x;
    if (i < n) { float d = deg[i]; deg[i] = (d > 0.0f) ? rsqrtf(d) : 0.0f; }
}

// --------------------------- h = embed_table[x] (float4 per thread) ---------
__global__ void hgcn_gather_h(const int* __restrict__ x,
                              const float* __restrict__ table,
                              float* __restrict__ h, int n) {
    size_t i = (size_t)blockIdx.x * 256 + threadIdx.x;   // over N*32 float4s
    size_t node = i >> 5;
    if (node >= (size_t)n) return;
    int q = (int)(i & 31);
    const float4* src = (const float4*)(table + (size_t)x[node] * DIMD);
    ((float4*)(h + node * DIMD))[q] = src[q];
}

// ------------- sem init: bias + self-loop message (norm = dinv^2) -----------
__global__ void hgcn_init_sem(float* __restrict__ sem, const float* __restrict__ hw,
                              const float* __restrict__ dinv,
                              const float* __restrict__ bias, int n) {
    size_t i = (size_t)blockIdx.x * 256 + threadIdx.x;   // over N*32 float4s
    size_t node = i >> 5;
    if (node >= (size_t)n) return;
    int q = (int)(i & 31);
    float di = dinv[node];
    float d2 = di * di;
    float4 hv = ((const float4*)(hw + node * DIMD))[q];
    float4 bv = ((const float4*)bias)[q];
    float4 r;
    r.x = bv.x + hv.x * d2; r.y = bv.y + hv.y * d2;
    r.z = bv.z + hv.z * d2; r.w = bv.w + hv.w * d2;
    ((float4*)(sem + node * DIMD))[q] = r;
}

// --------------------------- edge scatter (wave per edge) -------------------
// Lane owns 4 contiguous dims: one global_load_b128 + 4 f32 atomics per lane.
__global__ __launch_bounds__(256)
void hgcn_scatter(const float* __restrict__ hw, const int* __restrict__ src,
                  const int* __restrict__ dst, const float* __restrict__ dinv,
                  float* __restrict__ sem, int E) {
    int e = (int)(((size_t)blockIdx.x * 256 + threadIdx.x) >> 5);
    if (e >= E) return;
    int lane = threadIdx.x & 31;
    int s = src[e], d = dst[e];
    float norm = dinv[s] * dinv[d];
    float4 v = ((const float4*)(hw + (size_t)s * DIMD))[lane];
    float* sd = sem + (size_t)d * DIMD + lane * 4;
    atomicAdd(&sd[0], v.x * norm);
    atomicAdd(&sd[1], v.y * norm);
    atomicAdd(&sd[2], v.z * norm);
    atomicAdd(&sd[3], v.w * norm);
}

// ----- attention logit: logits[n,m] = tanh(t[n]+b1) . w2 (wave per node) ----
__global__ __launch_bounds__(256)
void hgcn_att_logits(const float* __restrict__ t, const float* __restrict__ b1,
                     const float* __restrict__ w2, float* __restrict__ logits,
                     int mIdx, int n) {
    int node = (int)(((size_t)blockIdx.x * 256 + threadIdx.x) >> 5);
    if (node >= n) return;
    int lane = threadIdx.x & 31;
    float4 tv = ((const float4*)(t + (size_t)node * DIMD))[lane];
    float4 bv = ((const float4*)b1)[lane];
    float4 wv = ((const float4*)w2)[lane];
    float s = tanhf(tv.x + bv.x) * wv.x + tanhf(tv.y + bv.y) * wv.y
            + tanhf(tv.z + bv.z) * wv.z + tanhf(tv.w + bv.w) * wv.w;
    #pragma unroll
    for (int off = 16; off > 0; off >>= 1) s += __shfl_xor(s, off, 32);
    if (lane == 0) logits[(size_t)node * 3 + mIdx] = s;
}

// ----- softmax over M=3, weighted sum; optional final log_softmax over D ----
__global__ __launch_bounds__(256)
void hgcn_combine(const float* __restrict__ sem, size_t semStride,
                  const float* __restrict__ logits, float* __restrict__ out,
                  int n, int lastLayer) {
    int node = (int)(((size_t)blockIdx.x * 256 + threadIdx.x) >> 5);
    if (node >= n) return;
    int lane = threadIdx.x & 31;

    float w0 = logits[(size_t)node * 3 + 0];
    float w1 = logits[(size_t)node * 3 + 1];
    float w2 = logits[(size_t)node * 3 + 2];
    float mx = fmaxf(w0, fmaxf(w1, w2));
    float e0 = expf(w0 - mx), e1 = expf(w1 - mx), e2 = expf(w2 - mx);
    float inv = 1.0f / (e0 + e1 + e2);
    float b0 = e0 * inv, b1 = e1 * inv, b2 = e2 * inv;

    size_t base = (size_t)node * DIMD;
    float4 z0 = ((const float4*)(sem + base))[lane];
    float4 z1 = ((const float4*)(sem + semStride + base))[lane];
    float4 z2 = ((const float4*)(sem + 2 * semStride + base))[lane];
    float hv[4];
    hv[0] = b0 * z0.x + b1 * z1.x + b2 * z2.x;
    hv[1] = b0 * z0.y + b1 * z1.y + b2 * z2.y;
    hv[2] = b0 * z0.z + b1 * z1.z + b2 * z2.z;
    hv[3] = b0 * z0.w + b1 * z1.w + b2 * z2.w;

    if (!lastLayer) {
        float4 r = {hv[0], hv[1], hv[2], hv[3]};
        ((float4*)(out + base))[lane] = r;
        return;
    }
    // log_softmax over D=128
    float m = fmaxf(fmaxf(hv[0], hv[1]), fmaxf(hv[2], hv[3]));
    #pragma unroll
    for (int off = 16; off > 0; off >>= 1) m = fmaxf(m, __shfl_xor(m, off, 32));
    float s = 0.0f;
    #pragma unroll
    for (int i = 0; i < 4; ++i) s += expf(hv[i] - m);
    #pragma unroll
    for (int off = 16; off > 0; off >>= 1) s += __shfl_xor(s, off, 32);
    float lse = m + logf(s);
    float4 r = {hv[0] - lse, hv[1] - lse, hv[2] - lse, hv[3] - lse};
    ((float4*)(out + base))[lane] = r;
}

// ---------------------------------------------------------------------------
extern "C" void kernel_launch(void* const* d_in, const int* in_sizes, int n_in,
                              void* d_out, int out_size, void* d_ws, size_t ws_size,
                              hipStream_t stream) {
    const int N = in_sizes[0];                 // 100000 nodes
    const int E = in_sizes[1] / 2;             // edges per meta-path
    const int L = 2, M = 3;

    const int*   x     = (const int*)d_in[0];
    const int*   ei[3] = { (const int*)d_in[1], (const int*)d_in[2], (const int*)d_in[3] };
    const float* table = (const float*)d_in[4];
    const float* W     = (const float*)d_in[5];   // [L,M,128,128]
    const float* Bb    = (const float*)d_in[6];   // [L,M,128]
    const float* aw1   = (const float*)d_in[7];   // [L,128,128]
    const float* ab1   = (const float*)d_in[8];   // [L,128]
    const float* aw2   = (const float*)d_in[9];   // [L,128,1]
    float* out = (float*)d_out;

    // ---- workspace carve-up (all fp32) ----
    const size_t feat = (size_t)N * DIMD;      // 12.8M floats
    char* ws = (char*)d_ws;
    float* h      = (float*)ws;                ws += feat * sizeof(float);
    float* hw     = (float*)ws;                ws += feat * sizeof(float);
    float* sem    = (float*)ws;                ws += 3 * feat * sizeof(float); // [M][N][128]
    float* dinv   = (float*)ws;                ws += (size_t)3 * N * sizeof(float);
    float* logits = (float*)ws;                ws += (size_t)3 * N * sizeof(float);
    (void)ws_size; (void)n_in; (void)out_size;

    const int TB = 256;
    const int nTiles = (N + 15) / 16;
    const int gNode  = (N + TB - 1) / TB;                      // per node
    const int gVec4  = (int)(((size_t)N * 32 + TB - 1) / TB);  // float4 per thread
    const int gWaveN = gVec4;                                  // wave per node
    const int gWaveE = (int)(((size_t)E * 32 + TB - 1) / TB);  // wave per edge
    const int gEdge  = (E + TB - 1) / TB;                      // thread per edge
    const int gGemm  = nTiles < 1024 ? nTiles : 1024;          // persistent tiles

    // ---- degrees (layer-invariant): deg = 1 (self loop) + indegree ----
    for (int m = 0; m < M; ++m) {
        float* dm = dinv + (size_t)m * N;
        hipLaunchKernelGGL(hgcn_init_deg,  dim3(gNode), dim3(TB), 0, stream, dm, N);
        hipLaunchKernelGGL(hgcn_count_deg, dim3(gEdge), dim3(TB), 0, stream, ei[m] + E, dm, E);
        hipLaunchKernelGGL(hgcn_rsqrt_deg, dim3(gNode), dim3(TB), 0, stream, dm, N);
    }

    // ---- h = embed_table[x] ----
    hipLaunchKernelGGL(hgcn_gather_h, dim3(gVec4), dim3(TB), 0, stream, x, table, h, N);

    for (int l = 0; l < L; ++l) {
        // GCN conv per meta-path
        for (int m = 0; m < M; ++m) {
            const float* Wlm = W  + ((size_t)(l * M + m)) * DIMD * DIMD;
            const float* blm = Bb + ((size_t)(l * M + m)) * DIMD;
            float* dm   = dinv + (size_t)m * N;
            float* semm = sem  + (size_t)m * feat;
            hipLaunchKernelGGL(hgcn_gemm_n128, dim3(gGemm), dim3(TB), 0, stream, h, Wlm, hw, N);
            hipLaunchKernelGGL(hgcn_init_sem,  dim3(gVec4), dim3(TB), 0, stream, semm, hw, dm, blm, N);
            hipLaunchKernelGGL(hgcn_scatter,   dim3(gWaveE), dim3(TB), 0, stream,
                               hw, ei[m], ei[m] + E, dm, semm, E);
        }
        // semantic attention logits
        for (int m = 0; m < M; ++m) {
            float* semm = sem + (size_t)m * feat;
            hipLaunchKernelGGL(hgcn_gemm_n128, dim3(gGemm), dim3(TB), 0, stream,
                               semm, aw1 + (size_t)l * DIMD * DIMD, hw, N);
            hipLaunchKernelGGL(hgcn_att_logits, dim3(gWaveN), dim3(TB), 0, stream,
                               hw, ab1 + (size_t)l * DIMD, aw2 + (size_t)l * DIMD,
                               logits, m, N);
        }
        // softmax over M + weighted sum (+ final log_softmax)
        const int last = (l == L - 1);
        hipLaunchKernelGGL(hgcn_combine, dim3(gWaveN), dim3(TB), 0, stream,
                           sem, feat, logits, last ? out : h, N, last);
    }
}